// FSNeuronDecoupled_80058190397932
// MI455X (gfx1250) — compile-verified
//
#include <hip/hip_runtime.h>
#include <stdint.h>

// FS-neuron decoupled forward, MI455X (gfx1250).
// Memory-bound elementwise op: 128 MB traffic -> ~5.5us floor @ 23.3 TB/s.
// Data path: saddr-form global_load_async_to_lds_b128 (ASYNCcnt, TH_LOAD_NT)
// double-buffered pipeline, ds_load_b128 from LDS, 4 VALU/step compute,
// non-temporal b128 stores. Ragged tail handled by one compact scalar path
// behind a block-uniform branch (no per-stage EXEC churn in the hot path).

typedef float v4f __attribute__((ext_vector_type(4)));

#define TPB 256
#define STAGES 4
#define ELEMS_PER_BLOCK (TPB * 4 * STAGES)   // 4096 floats per block
#define STAGE_BYTES (TPB * 4 * 4)            // 4096 bytes per stage

// c[t] = base(t)^-(t+1), base = b0 for t<4, b1 for t>=4 (args are 1/base).
__device__ __forceinline__ void coeffs_pow(float ib0, float ib1, float* c) {
  c[0] = ib0;
  c[1] = c[0] * ib0;
  c[2] = c[1] * ib0;
  c[3] = c[2] * ib0;
  float b2 = ib1 * ib1;
  float b4 = b2 * b2;
  c[4] = b4 * ib1;          // ib1^5
  c[5] = c[4] * ib1;
  c[6] = c[5] * ib1;
  c[7] = c[6] * ib1;
}

__device__ __forceinline__ float fs_neuron(float v, const float* th,
                                           const float* hh, const float* dd) {
  float acc = 0.0f;
#pragma unroll
  for (int t = 0; t < 8; ++t) {
    float s = (v > th[t]) ? 1.0f : 0.0f;       // heaviside(v - theta_t)
    v   = __builtin_fmaf(-hh[t], s, v);        // v -= h_t * s
    acc = __builtin_fmaf( dd[t], s, acc);      // acc += d_t * s
  }
  return acc;
}

// Async copy 16 bytes/lane global->LDS, saddr addressing:
//   mem = SGPR_base + voff + imm_offset ; lds = lds_addr (workgroup-relative).
// Non-temporal: the input stream is touched exactly once.
__device__ __forceinline__ void async_load_b128(uint32_t lds_addr, uint32_t voff,
                                                const float* sbase, int ioff) {
  asm volatile("global_load_async_to_lds_b128 %0, %1, %2 offset:%3 th:TH_LOAD_NT"
               :: "v"(lds_addr), "v"(voff), "s"(sbase), "i"(ioff)
               : "memory");
}

__global__ __launch_bounds__(TPB) void fsneuron_kernel(
    const float* __restrict__ x,
    const float* __restrict__ d,
    const float* __restrict__ h,
    const float* __restrict__ theta,
    float* __restrict__ out,
    long long total)
{
  __shared__ v4f buf[2][TPB];   // 8 KB double buffer, wave-private slots

  // Per-step coefficients from the six wave-uniform scalars (s_load path).
  float th[8], hh[8], dd[8];
  coeffs_pow(1.0f / theta[0], 1.0f / theta[1], th);
  coeffs_pow(1.0f / h[0],     1.0f / h[1],     hh);
  coeffs_pow(1.0f / d[0],     1.0f / d[1],     dd);

  const int tid = threadIdx.x;
  const long long blockBase = (long long)blockIdx.x * ELEMS_PER_BLOCK;

  if (blockBase + ELEMS_PER_BLOCK <= total) {
    // ---- hot path: full block, no per-lane guards, EXEC stays all-ones ----
    const float* xb = x + blockBase;       // block-uniform -> SGPR pair
    float*       ob = out + blockBase;

    // Low 32 bits of a generic pointer into __shared__ == workgroup-relative
    // LDS byte address; exactly what the async-load VDST operand wants.
    uint32_t lds0 = (uint32_t)(uintptr_t)&buf[0][tid];
    uint32_t lds1 = (uint32_t)(uintptr_t)&buf[1][tid];
    uint32_t voff = (uint32_t)tid * 16u;   // per-lane byte offset

    async_load_b128(lds0, voff, xb, 0);    // stage 0 in flight

#pragma unroll
    for (int s = 0; s < STAGES; ++s) {
      if (s + 1 < STAGES) {
        async_load_b128((s & 1) ? lds0 : lds1, voff, xb, (s + 1) * STAGE_BYTES);
        // Two async loads outstanding; they complete in order -> wait <=1
        // guarantees stage s is resident in LDS.
        asm volatile("s_wait_asynccnt 0x1" ::: "memory");
      } else {
        asm volatile("s_wait_asynccnt 0x0" ::: "memory");
      }

      v4f xv = buf[s & 1][tid];            // ds_load_b128
      v4f yv;
      yv.x = fs_neuron(xv.x, th, hh, dd);
      yv.y = fs_neuron(xv.y, th, hh, dd);
      yv.z = fs_neuron(xv.z, th, hh, dd);
      yv.w = fs_neuron(xv.w, th, hh, dd);
      __builtin_nontemporal_store(yv, (v4f*)(ob + s * (TPB * 4) + tid * 4));
    }
  } else {
    // ---- ragged tail block (never taken for the harness shape) ----
    // Single compact scalar loop; fs_neuron instantiated once here.
#pragma clang loop unroll(disable)
    for (int s = 0; s < STAGES * 4; ++s) {
      long long e = blockBase + (long long)s * TPB + tid;
      if (e < total) out[e] = fs_neuron(x[e], th, hh, dd);
    }
  }
}

extern "C" void kernel_launch(void* const* d_in, const int* in_sizes, int n_in,
                              void* d_out, int out_size, void* d_ws, size_t ws_size,
                              hipStream_t stream) {
  const float* x     = (const float*)d_in[0];
  const float* d     = (const float*)d_in[1];
  const float* h     = (const float*)d_in[2];
  const float* theta = (const float*)d_in[3];
  float* out = (float*)d_out;

  const long long total = (long long)in_sizes[0];       // 16,777,216
  const int blocks = (int)((total + ELEMS_PER_BLOCK - 1) / ELEMS_PER_BLOCK);

  fsneuron_kernel<<<blocks, TPB, 0, stream>>>(x, d, h, theta, out, total);
}